// SigLayer_52544629899941
// MI455X (gfx1250) — compile-verified
//
#include <hip/hip_runtime.h>

typedef __attribute__((ext_vector_type(2))) float v2f;
typedef __attribute__((ext_vector_type(8))) float v8f;
typedef __attribute__((ext_vector_type(4))) int   v4i;

#define BATCH 64
#define LEN   256
#define CH    16
#define TREAL 255          // real increments
#define TK    256          // padded (dx[255] = 0; zero increment is a no-op in Chen)
#define CHUNK 32
#define NCHUNK (TK / CHUNK)
#define OUT_STRIDE (16 + 256 + 4096)   // 4368 floats per batch

// gfx1250 async global->LDS copy path; fall back to plain loads if the
// builtins aren't available in this toolchain.
#define TRY_ASYNC 1
#if TRY_ASYNC && __has_builtin(__builtin_amdgcn_global_load_async_to_lds_b128) && \
    __has_builtin(__builtin_amdgcn_s_wait_asynccnt)
#define USE_ASYNC 1
#else
#define USE_ASYNC 0
#endif

// One workgroup per batch. 256 threads = 8 wave32.
//  - dx -> LDS (async copy of raw path + diff, or direct loads)
//  - 8 chunks of 32 steps:
//      scan: thread (i,j) carries running P1[i] and A2[i,j] in registers,
//            writes R[t][(i,j)] (and Q[t][i] for j==0) into double-buffered LDS
//      wmma: wave w accumulates S3 M-tiles i=2w,2w+1; every wave redundantly
//            accumulates the single S2 tile (uniform 3 WMMAs/K-step, no EXEC
//            toggling); K=4 fp32 WMMA, accumulators live across all chunks
//  - store S1 | S2 | S3
__global__ __launch_bounds__(256)
void sig_depth3_kernel(const float* __restrict__ path, float* __restrict__ out) {
    __shared__ float dxs[TK][CH];            // 16 KB
#if USE_ASYNC
    __shared__ float pbuf[LEN][CH];          // 16 KB raw path tile
#endif
    __shared__ float Rbuf[2][CHUNK][256];    // 64 KB (double-buffered)
    __shared__ float Qbuf[2][CHUNK][CH];     //  4 KB (double-buffered)

    const int b    = blockIdx.x;
    const int tid  = threadIdx.x;            // 0..255
    const int lane = tid & 31;
    const int wave = tid >> 5;               // 0..7
    const int m    = lane & 15;              // matrix row/col lane index
    const int half = lane >> 4;              // 0 or 1

    const float* P = path + (size_t)b * LEN * CH;

    // gfx1250 prefetch path (lowers to global_prefetch_b8)
    __builtin_prefetch(P + tid * 16, 0, 1);

    // ---- Stage the path / increments into LDS ----
#if USE_ASYNC
    {
        typedef __attribute__((address_space(1))) v4i* gv4;
        typedef __attribute__((address_space(3))) v4i* lv4;
        #pragma unroll
        for (int it = 0; it < (LEN * CH) / (256 * 4); ++it) {   // 4 x b128 per thread
            int e = (it * 256 + tid) * 4;                        // float index
            __builtin_amdgcn_global_load_async_to_lds_b128(
                (gv4)(P + e), (lv4)(&pbuf[0][0] + e), 0, 0);
        }
        __builtin_amdgcn_s_wait_asynccnt(0);
    }
    __syncthreads();
    for (int idx = tid; idx < TREAL * CH; idx += 256) {
        int t = idx >> 4, c2 = idx & 15;
        dxs[t][c2] = pbuf[t + 1][c2] - pbuf[t][c2];
    }
#else
    for (int idx = tid; idx < TREAL * CH; idx += 256) {
        int t = idx >> 4, c2 = idx & 15;
        dxs[t][c2] = P[(t + 1) * CH + c2] - P[t * CH + c2];
    }
#endif
    if (tid < CH) dxs[TREAL][tid] = 0.0f;    // zero-pad step 255
    __syncthreads();

    // Per-thread scan state: thread owns pair (i,j)
    const int oi = tid >> 4;
    const int oj = tid & 15;
    float a2  = 0.0f;   // running level-2 A2[i,j]
    float p1i = 0.0f;   // running level-1 prefix P1[i]

    // WMMA accumulators (held across all chunks)
    v8f acc0 = {};      // S3 tile i = 2*wave
    v8f acc1 = {};      // S3 tile i = 2*wave + 1
    v8f acc2 = {};      // S2 (computed redundantly by every wave)

    const int tileA = 2 * wave;
    const int tileB = 2 * wave + 1;

    for (int c = 0; c < NCHUNK; ++c) {
        const int buf   = c & 1;
        const int tbase = c * CHUNK;

        // ---- scan 32 steps: R (and Q) into LDS ----
        for (int tt = 0; tt < CHUNK; ++tt) {
            int t = tbase + tt;
            float vi = dxs[t][oi];
            float vj = dxs[t][oj];
            Rbuf[buf][tt][tid] = a2 + 0.5f * p1i * vj + (1.0f / 6.0f) * vi * vj;
            if (oj == 0) Qbuf[buf][tt][oi] = p1i + 0.5f * vi;
            a2  += p1i * vj + 0.5f * vi * vj;
            p1i += vi;
        }
        __syncthreads();   // single barrier per chunk (double-buffered R/Q)

        // ---- WMMA accumulation over this chunk, K=4 per instruction ----
        for (int k0 = 0; k0 < CHUNK; k0 += 4) {
            const int ka = k0 + 2 * half;       // local K pair base
            const int ta = tbase + ka;          // global t for dx loads

            // B: 4x16 fp32 tile of DX (lanes 0-15: K rows 0,1; 16-31: rows 2,3)
            v2f Bm;
            Bm.x = dxs[ta + 0][m];
            Bm.y = dxs[ta + 1][m];

            // A: 16x4 fp32 tiles of R (rows = j for fixed i = tile)
            v2f A0;
            A0.x = Rbuf[buf][ka + 0][tileA * 16 + m];
            A0.y = Rbuf[buf][ka + 1][tileA * 16 + m];
            v2f A1;
            A1.x = Rbuf[buf][ka + 0][tileB * 16 + m];
            A1.y = Rbuf[buf][ka + 1][tileB * 16 + m];
            // A: 16x4 fp32 tile of Q for S2
            v2f Aq;
            Aq.x = Qbuf[buf][ka + 0][m];
            Aq.y = Qbuf[buf][ka + 1][m];

            acc0 = __builtin_amdgcn_wmma_f32_16x16x4_f32(
                false, A0, false, Bm, (short)0, acc0, false, false);
            acc1 = __builtin_amdgcn_wmma_f32_16x16x4_f32(
                false, A1, false, Bm, (short)0, acc1, false, false);
            acc2 = __builtin_amdgcn_wmma_f32_16x16x4_f32(
                false, Aq, false, Bm, (short)0, acc2, false, false);
        }
    }

    // ---- store results ----
    float* ob = out + (size_t)b * OUT_STRIDE;

    // Level 1: thread (i, j==0) holds the full channel-i sum in p1i
    if (oj == 0) ob[oi] = p1i;

    // Level 3: D layout: lane -> n=k (m), VGPR r -> row j = r + 8*half
    {
        float* o3 = ob + 16 + 256;
        #pragma unroll
        for (int r = 0; r < 8; ++r) {
            int j = r + 8 * half;
            o3[tileA * 256 + j * 16 + m] = acc0[r];
            o3[tileB * 256 + j * 16 + m] = acc1[r];
        }
    }

    // Level 2: only wave 0 stores its (redundant) copy
    if (wave == 0) {
        float* o2 = ob + 16;
        #pragma unroll
        for (int r = 0; r < 8; ++r) {
            int i = r + 8 * half;
            o2[i * 16 + m] = acc2[r];
        }
    }
}

extern "C" void kernel_launch(void* const* d_in, const int* in_sizes, int n_in,
                              void* d_out, int out_size, void* d_ws, size_t ws_size,
                              hipStream_t stream) {
    (void)in_sizes; (void)n_in; (void)d_ws; (void)ws_size; (void)out_size;
    const float* path = (const float*)d_in[0];
    float* out = (float*)d_out;
    sig_depth3_kernel<<<BATCH, 256, 0, stream>>>(path, out);
}